// RMSENormalizedLoss_91164975825342
// MI455X (gfx1250) — compile-verified
//
#include <hip/hip_runtime.h>
#include <math.h>

typedef __attribute__((ext_vector_type(2))) float v2f;
typedef __attribute__((ext_vector_type(4))) float v4f;
typedef __attribute__((ext_vector_type(8))) float v8f;

#define BLOCK_THREADS 256
#define WAVES_PER_BLOCK (BLOCK_THREADS / 32)

// One block per row. 256 threads * float4 = 1024 columns per iteration.
// WMMA f32 16x16x4 with B == ones acts as the reduction adder:
// D[m][n] = sum_k A[m][k] + C[m][n]; every column of D is identical.
__global__ __launch_bounds__(BLOCK_THREADS)
void rmse_partial_kernel(const float* __restrict__ pred,
                         const float* __restrict__ act,
                         const int* __restrict__ indexes,
                         const int* __restrict__ starting_indexes,
                         float* __restrict__ part_sum,
                         unsigned int* __restrict__ part_cnt,
                         int T) {
    const int row  = blockIdx.x;
    const int tid  = threadIdx.x;
    const int lane = tid & 31;
    const int wave = tid >> 5;

    const int start = starting_indexes[indexes[row]];   // uniform per block

    const float* __restrict__ prow = pred + (size_t)row * (size_t)T;
    const float* __restrict__ arow = act  + (size_t)row * (size_t)T;

    v8f c = {0.f, 0.f, 0.f, 0.f, 0.f, 0.f, 0.f, 0.f};
    const v2f ones = {1.0f, 1.0f};

    const int stride = BLOCK_THREADS * 4;               // 1024 columns/iter
    for (int base = 0; base < T; base += stride) {
        const int col = base + tid * 4;
        v4f p = *(const v4f*)(prow + col);              // global_load_b128
        v4f a = *(const v4f*)(arow + col);              // global_load_b128

        // masked squared difference (data-divergent only; EXEC stays all-1s)
        float s0 = p.x - a.x; s0 = (col + 0 >= start) ? s0 : 0.f; s0 *= s0;
        float s1 = p.y - a.y; s1 = (col + 1 >= start) ? s1 : 0.f; s1 *= s1;
        float s2 = p.z - a.z; s2 = (col + 2 >= start) ? s2 : 0.f; s2 *= s2;
        float s3 = p.w - a.w; s3 = (col + 3 >= start) ? s3 : 0.f; s3 *= s3;

        v2f a0 = {s0, s1};
        v2f a1 = {s2, s3};
        // D = A*ones + C  -> accumulate tile row-sums into c
        c = __builtin_amdgcn_wmma_f32_16x16x4_f32(
                /*neg_a=*/false, a0, /*neg_b=*/false, ones,
                /*c_mod=*/(short)0, c, /*reuse_a=*/false, /*reuse_b=*/false);
        c = __builtin_amdgcn_wmma_f32_16x16x4_f32(
                false, a1, false, ones, (short)0, c, false, false);
    }

    // Per-lane sum of the 8 accumulator VGPRs. Every column of D is
    // bit-identical, so the full 32-lane xor-reduce gives exactly 16x the
    // wave total; dividing by 16 is exact in binary FP.
    float s = c[0] + c[1] + c[2] + c[3] + c[4] + c[5] + c[6] + c[7];
    #pragma unroll
    for (int m = 16; m >= 1; m >>= 1)
        s += __shfl_xor(s, m, 32);
    s *= (1.0f / 16.0f);

    __shared__ float wave_sum[WAVES_PER_BLOCK];
    if (lane == 0) wave_sum[wave] = s;
    __syncthreads();

    if (tid == 0) {
        float t = 0.f;
        #pragma unroll
        for (int w = 0; w < WAVES_PER_BLOCK; ++w) t += wave_sum[w];
        part_sum[row] = t;
        part_cnt[row] = (unsigned int)(T - start);      // exact masked count
    }
}

__global__ __launch_bounds__(BLOCK_THREADS)
void rmse_finalize_kernel(const float* __restrict__ part_sum,
                          const unsigned int* __restrict__ part_cnt,
                          int nblocks,
                          float* __restrict__ out) {
    __shared__ double dsum[BLOCK_THREADS];
    __shared__ unsigned long long csum[BLOCK_THREADS];

    double s = 0.0;
    unsigned long long c = 0ull;
    for (int i = threadIdx.x; i < nblocks; i += BLOCK_THREADS) {
        s += (double)part_sum[i];
        c += (unsigned long long)part_cnt[i];
    }
    dsum[threadIdx.x] = s;
    csum[threadIdx.x] = c;
    __syncthreads();

    for (int off = BLOCK_THREADS / 2; off > 0; off >>= 1) {
        if (threadIdx.x < off) {
            dsum[threadIdx.x] += dsum[threadIdx.x + off];
            csum[threadIdx.x] += csum[threadIdx.x + off];
        }
        __syncthreads();
    }

    if (threadIdx.x == 0)
        out[0] = (float)sqrt(dsum[0] / (double)csum[0]);
}

extern "C" void kernel_launch(void* const* d_in, const int* in_sizes, int n_in,
                              void* d_out, int out_size, void* d_ws, size_t ws_size,
                              hipStream_t stream) {
    const float* predictions      = (const float*)d_in[0];
    const float* actuals          = (const float*)d_in[1];
    const int*   indexes          = (const int*)d_in[2];
    const int*   starting_indexes = (const int*)d_in[3];

    const int B = in_sizes[2];            // number of rows (len of indexes)
    const int T = in_sizes[0] / B;        // columns per row

    float*        part_sum = (float*)d_ws;
    unsigned int* part_cnt = (unsigned int*)((char*)d_ws + (size_t)B * sizeof(float));

    rmse_partial_kernel<<<B, BLOCK_THREADS, 0, stream>>>(
        predictions, actuals, indexes, starting_indexes, part_sum, part_cnt, T);

    rmse_finalize_kernel<<<1, BLOCK_THREADS, 0, stream>>>(
        part_sum, part_cnt, B, (float*)d_out);
}